// DigitCap_4071628997245
// MI455X (gfx1250) — compile-verified
//
#include <hip/hip_runtime.h>
#include <math.h>

typedef __attribute__((ext_vector_type(2))) float v2f;
typedef __attribute__((ext_vector_type(4))) float v4f;
typedef __attribute__((ext_vector_type(8))) float v8f;

#define B_  256
#define N_  8
#define P_  1152
#define L_  10
#define O_  16
#define K_  (P_ * N_)     // 9216
#define M_  (L_ * O_)     // 160
#define KCHUNKS 8
#define KPER (K_ / KCHUNKS)  // 1152

// ---------------------------------------------------------------------------
// Pass 1: LDS-tiled transpose+permute into a PAIR-INTERLEAVED B layout:
//   ut2[(k>>1)*(2*B) + b*2 + (k&1)] = x[b][n][p],  k = p*8 + n
// With k = p*8 + (2m+j), the pair (k even, k odd) = n-planes (2m, 2m+1) of the
// same p.  Each block stages two n-planes of a 32(b) x 32(p) tile in LDS:
// reads coalesced along p, writes one aligned float2 per lane (coalesced).
// This makes every B fragment in the GEMM a single 8-byte b64 load.
// ---------------------------------------------------------------------------
__global__ void xpose_kernel(const float* __restrict__ x, float* __restrict__ ut2) {
    __shared__ float tA[32][33];
    __shared__ float tB[32][33];
    const int p0 = blockIdx.x * 32;       // 0..1151 step 32
    const int b0 = blockIdx.y * 32;       // 0..255  step 32
    const int m  = blockIdx.z;            // 0..3 : n pair (n = 2m, 2m+1)
    const int tx = threadIdx.x;           // 0..31
    const int ty = threadIdx.y;           // 0..7

    #pragma unroll
    for (int yy = ty; yy < 32; yy += 8) {
        const float* row = x + (size_t)(b0 + yy) * K_ + (2 * m) * P_ + p0 + tx;
        tA[yy][tx] = row[0];    // n = 2m
        tB[yy][tx] = row[P_];   // n = 2m+1
    }
    __syncthreads();

    #pragma unroll
    for (int yy = ty; yy < 32; yy += 8) {
        v2f v;
        v.x = tA[tx][yy];
        v.y = tB[tx][yy];
        // pair index k2 = p*4 + m ; element j = k&1 = n-2m
        *(v2f*)(ut2 + ((size_t)(p0 + yy) * 4 + m) * (2 * B_) + (size_t)(b0 + tx) * 2) = v;
    }
}

// ---------------------------------------------------------------------------
// Pass 2: WMMA GEMM  s[b, l*16+o] = sum_k A[m,k] * Bm[k,b]
//   A[m,k]  = W[p, l, o, n]  (m = l*16+o, k = p*8+n)  -> float2 loads from W
//   Bm pairs from ut2                                 -> one b64 per fragment
// K stepped by 8: both half-waves share p, so the two A loads of a double-step
// are constant offsets off one pointer; all pointers bump by constants
// (A: +1280 floats, B: +2048 floats) -> strength-reduced addressing.
// One wave per 16x16 tile per K-chunk; v_wmma_f32_16x16x4_f32 accumulate.
// ---------------------------------------------------------------------------
__global__ void gemm_wmma_kernel(const float* __restrict__ W,
                                 const float* __restrict__ ut2,
                                 float* __restrict__ s_part) {
    const int btile = blockIdx.x;     // 0..15  (batch tile)
    const int ltile = blockIdx.y;     // 0..9   (label = M tile)
    const int chunk = blockIdx.z;     // 0..7   (K split)
    const int lane  = threadIdx.x;    // 0..31, wave32, EXEC all ones
    const int half  = lane >> 4;      // 0: K+{0,1}, 1: K+{2,3}
    const int lmod  = lane & 15;
    const int bcol  = btile * 16 + lmod;

    const int k_begin = chunk * KPER;               // multiple of 8

    // A: lane's row m=lmod of ltile at p = k/8, n = k%8:
    //    addr = W + p*1280 + ltile*128 + m*8 + n
    const float* aPtr = W + ltile * (O_ * N_) + lmod * N_
                          + (size_t)(k_begin >> 3) * (P_ == 0 ? 0 : L_ * O_ * N_)
                          + half * 2;
    // B: pair row k2 = k/2, lane column bcol
    const float* bPtr = ut2 + ((size_t)(k_begin >> 1) + half) * (2 * B_) + bcol * 2;

    v8f c = {};
    #pragma unroll 4
    for (int it = 0; it < KPER / 8; ++it) {
        v2f a01 = *(const v2f*)(aPtr);          // A[m, k0+h2 .. +1]
        v2f a23 = *(const v2f*)(aPtr + 4);      // A[m, k0+4+h2 .. +1]
        v2f b01 = *(const v2f*)(bPtr);          // B[k0+h2 .. +1, b]
        v2f b23 = *(const v2f*)(bPtr + 4 * B_); // B[k0+4+h2 .. +1, b]
        c = __builtin_amdgcn_wmma_f32_16x16x4_f32(
                false, a01, false, b01, (short)0, c, false, false);
        c = __builtin_amdgcn_wmma_f32_16x16x4_f32(
                false, a23, false, b23, (short)0, c, false, false);
        aPtr += L_ * O_ * N_;   // +1280 floats: next p
        bPtr += 8 * B_;         // +2048 floats: next K-octet
    }

    // C/D layout: VGPR r -> row m = half*8 + r, col = lmod. Rows contiguous in
    // the (b, l, o) output (o stride 1) -> two aligned float4 stores.
    float* dst = s_part + (size_t)chunk * (B_ * M_)
                        + (size_t)bcol * M_ + ltile * 16 + half * 8;
    v4f lo = {c[0], c[1], c[2], c[3]};
    v4f hi = {c[4], c[5], c[6], c[7]};
    *(v4f*)(dst)     = lo;
    *(v4f*)(dst + 4) = hi;
}

// ---------------------------------------------------------------------------
// Pass 3: reduce the 8 K-chunk partials and apply squash.
// Faithful to the reference: norm taken over axis 2 == the label axis (L).
// Routing iterations are a no-op (softmax over a size-1 axis == 1.0), so the
// output is just squash(sum_p u_hat).
// ---------------------------------------------------------------------------
__global__ void squash_kernel(const float* __restrict__ s_part,
                              float* __restrict__ out) {
    const int t = blockIdx.x * blockDim.x + threadIdx.x;   // 0 .. B_*O_-1
    if (t >= B_ * O_) return;
    const int b = t >> 4;
    const int o = t & 15;

    float sv[L_];
    float msq = 0.0f;
    #pragma unroll
    for (int l = 0; l < L_; ++l) {
        float acc = 0.0f;
        #pragma unroll
        for (int c = 0; c < KCHUNKS; ++c)
            acc += s_part[(size_t)c * (B_ * M_) + (size_t)b * M_ + l * 16 + o];
        sv[l] = acc;
        msq += acc * acc;
    }
    const float mag   = sqrtf(msq);
    const float scale = msq / ((1.0f + msq) * mag);
    #pragma unroll
    for (int l = 0; l < L_; ++l)
        out[(size_t)b * M_ + l * 16 + o] = scale * sv[l];
}

// ---------------------------------------------------------------------------
extern "C" void kernel_launch(void* const* d_in, const int* in_sizes, int n_in,
                              void* d_out, int out_size, void* d_ws, size_t ws_size,
                              hipStream_t stream) {
    const float* x = (const float*)d_in[0];   // (256, 8, 1152) fp32
    const float* W = (const float*)d_in[1];   // (1, 1152, 10, 16, 8) fp32
    float* out = (float*)d_out;               // (256, 10, 16) fp32

    // Workspace: ut2 (K_*B_ floats = 9.44 MB) | s_part (8*B_*M_ floats = 1.31 MB)
    float* ut2    = (float*)d_ws;
    float* s_part = ut2 + (size_t)K_ * B_;

    // Pass 1: transpose x -> pair-interleaved ut2
    xpose_kernel<<<dim3(P_ / 32, B_ / 32, N_ / 2), dim3(32, 8), 0, stream>>>(x, ut2);

    // Pass 2: WMMA GEMM with 8-way K split (1280 waves)
    gemm_wmma_kernel<<<dim3(B_ / 16, L_, KCHUNKS), 32, 0, stream>>>(W, ut2, s_part);

    // Pass 3: chunk reduction + squash
    squash_kernel<<<dim3((B_ * O_ + 255) / 256), 256, 0, stream>>>(s_part, out);
}